// VolumeSDFRenderer_33913061769934
// MI455X (gfx1250) — compile-verified
//
#include <hip/hip_runtime.h>
#include <stdint.h>

// VolumeSDF renderer: memory-bound streaming scan + weighted color reduction.
// One wave32 per ray (N_PTS=128 -> 4 points/lane).
// gfx1250 paths: global_load_async_to_lds_b128 (ASYNCcnt DMA) + s_wait_asynccnt,
// wave32 shuffle scan/reduction.

#define N_PTS          128
#define RAYS_PER_BLOCK 8            // 8 waves * 32 lanes = 256 threads
#define SDF_SCALE      150.0f
#define FAR_DELTA      1e10f

__global__ __launch_bounds__(256)
void VolumeSDFRenderer_kernel(const float* __restrict__ distance,
                              const float* __restrict__ color,
                              const float* __restrict__ depth,
                              float* __restrict__ out,
                              int n_rays)
{
    // Color staging buffer: 8 rays * 128 pts * 3 ch * 4B = 12 KB
    __shared__ __align__(16) float cbuf[RAYS_PER_BLOCK * N_PTS * 3];

    const int lane = threadIdx.x & 31;
    const int wave = threadIdx.x >> 5;
    const int ray  = blockIdx.x * RAYS_PER_BLOCK + wave;
    if (ray >= n_rays) return;           // wave-uniform; grid divides exactly anyway

    // ------------------------------------------------------------------
    // Kick off async DMA of this ray's color row (1536 B) into LDS.
    // GVS addressing: saddr(64-bit SGPR base) + vaddr(32-bit per-lane offset).
    // Each b128 op moves 32 lanes * 16 B = 512 B -> 3 ops per ray.
    // Issued before any compute so the DMA overlaps the scan math.
    // ------------------------------------------------------------------
    const unsigned lds_base = (unsigned)(uintptr_t)(&cbuf[wave * N_PTS * 3]);
    const unsigned lofs     = lds_base + (unsigned)(lane * 16);
    const int      gofs     = ray * (N_PTS * 3 * 4) + lane * 16;

    asm volatile("global_load_async_to_lds_b128 %0, %1, %2"
                 :: "v"(lofs),         "v"(gofs),        "s"(color) : "memory");
    asm volatile("global_load_async_to_lds_b128 %0, %1, %2"
                 :: "v"(lofs + 512u),  "v"(gofs + 512),  "s"(color) : "memory");
    asm volatile("global_load_async_to_lds_b128 %0, %1, %2"
                 :: "v"(lofs + 1024u), "v"(gofs + 1024), "s"(color) : "memory");

    // ------------------------------------------------------------------
    // Coalesced b128 loads of distance & depth (512 B per wave each).
    // ------------------------------------------------------------------
    const int p0 = lane * 4;
    const float4 d4 = *(const float4*)(distance + (size_t)ray * N_PTS + p0);
    const float4 z4 = *(const float4*)(depth    + (size_t)ray * N_PTS + p0);

    // deltas: z[p+1]-z[p]; last point gets FAR_DELTA
    const float znext = __shfl_down(z4.x, 1, 32);           // next lane's first depth
    const float dl0 = z4.y - z4.x;
    const float dl1 = z4.z - z4.y;
    const float dl2 = z4.w - z4.z;
    const float dl3 = (lane == 31) ? FAR_DELTA : (znext - z4.w);

    // density = 150*e/(1+e)^2, e = exp(-150*d)   (same expression order as ref)
    const float e0 = __expf(-SDF_SCALE * d4.x);
    const float e1 = __expf(-SDF_SCALE * d4.y);
    const float e2 = __expf(-SDF_SCALE * d4.z);
    const float e3 = __expf(-SDF_SCALE * d4.w);
    const float x0 = dl0 * ((SDF_SCALE * e0) / ((1.0f + e0) * (1.0f + e0)));
    const float x1 = dl1 * ((SDF_SCALE * e1) / ((1.0f + e1) * (1.0f + e1)));
    const float x2 = dl2 * ((SDF_SCALE * e2) / ((1.0f + e2) * (1.0f + e2)));
    const float x3 = dl3 * ((SDF_SCALE * e3) / ((1.0f + e3) * (1.0f + e3)));

    // in-lane inclusive prefix over the 4 local points
    const float c0 = x0;
    const float c1 = c0 + x1;
    const float c2 = c1 + x2;
    const float c3 = c2 + x3;

    // wave32 inclusive scan of lane totals -> exclusive base per lane
    const float tot = c3;
    float scan = tot;
    #pragma unroll
    for (int ofs = 1; ofs < 32; ofs <<= 1) {
        const float n = __shfl_up(scan, ofs, 32);
        if (lane >= ofs) scan += n;
    }
    const float base = scan - tot;       // exclusive cumsum entering this lane

    // T_i = exp(-exclusive_cumsum_i), w_i = T_i * (1 - exp(-x_i))
    const float w0 = __expf(-base)        * (1.0f - __expf(-x0));
    const float w1 = __expf(-(base + c0)) * (1.0f - __expf(-x1));
    const float w2 = __expf(-(base + c1)) * (1.0f - __expf(-x2));
    const float w3 = __expf(-(base + c2)) * (1.0f - __expf(-x3));

    // ------------------------------------------------------------------
    // Retire the async DMA, then consume color from LDS.
    // ------------------------------------------------------------------
    asm volatile("s_wait_asynccnt 0x0" ::: "memory");

    const float* cl = &cbuf[wave * N_PTS * 3 + p0 * 3];     // 12 floats for 4 points
    const float4 ca = *(const float4*)(cl + 0);   // p0.r p0.g p0.b p1.r
    const float4 cb = *(const float4*)(cl + 4);   // p1.g p1.b p2.r p2.g
    const float4 cc = *(const float4*)(cl + 8);   // p2.b p3.r p3.g p3.b

    float r = w0 * ca.x + w1 * ca.w + w2 * cb.z + w3 * cc.y;
    float g = w0 * ca.y + w1 * cb.x + w2 * cb.w + w3 * cc.z;
    float b = w0 * ca.z + w1 * cb.y + w2 * cc.x + w3 * cc.w;

    // cross-lane reduction of the 3 channels
    #pragma unroll
    for (int ofs = 16; ofs > 0; ofs >>= 1) {
        r += __shfl_xor(r, ofs, 32);
        g += __shfl_xor(g, ofs, 32);
        b += __shfl_xor(b, ofs, 32);
    }

    if (lane == 0) {
        out[ray * 3 + 0] = r;
        out[ray * 3 + 1] = g;
        out[ray * 3 + 2] = b;
    }
}

extern "C" void kernel_launch(void* const* d_in, const int* in_sizes, int n_in,
                              void* d_out, int out_size, void* d_ws, size_t ws_size,
                              hipStream_t stream) {
    const float* distance = (const float*)d_in[0];   // [R, 128]
    const float* color    = (const float*)d_in[1];   // [R, 128, 3]
    const float* depth    = (const float*)d_in[2];   // [R, 128]
    float* out = (float*)d_out;                      // [R, 3]

    const int n_rays = in_sizes[0] / N_PTS;
    const int blocks = (n_rays + RAYS_PER_BLOCK - 1) / RAYS_PER_BLOCK;

    hipLaunchKernelGGL(VolumeSDFRenderer_kernel, dim3(blocks), dim3(256), 0, stream,
                       distance, color, depth, out, n_rays);
}